// DenseCapsLayer_30726196035850
// MI455X (gfx1250) — compile-verified
//
#include <hip/hip_runtime.h>
#include <math.h>
#include <stdint.h>

typedef float v2f __attribute__((ext_vector_type(2)));
typedef float v8f __attribute__((ext_vector_type(8)));
typedef uint32_t u32x4 __attribute__((ext_vector_type(4)));
typedef uint32_t u32x8 __attribute__((ext_vector_type(8)));

#define HW   256   // oh*ow
#define NA   32
#define NB   32
#define PS   16    // P*P
#define NITR 3
#define LTS  258   // padded stride for logitT[b][h]
#define ETS  258   // padded stride for et[b][h]
#define PTS  258   // padded stride for poseT[e][h]

// One block per (n, a). 256 threads = 8 wave32.
__global__ __launch_bounds__(256) void caps_route_kernel(const float* __restrict__ x,
                                                         const float* __restrict__ w,
                                                         float* __restrict__ out) {
  const int tid  = threadIdx.x;
  const int lane = tid & 31;
  const int wave = tid >> 5;
  const int lm   = lane & 15;   // row/col-in-tile index
  const int lh   = lane >> 4;   // K-half selector for f32 WMMA fragments
  const int blk  = blockIdx.x;
  const int n    = blk >> 5;
  const int a    = blk & 31;

  __shared__ float pose_s[HW * PS];      // 16 KB : pose[h][e]   (TDM destination)
  __shared__ float poseT_s[PS * PTS];    // poseT[e][h] (padded) for GEMM1 B-frags
  __shared__ float logitT_s[NB * LTS];   // logits[b][h] (padded)
  __shared__ float et_s[NB * ETS];       // exp(logit-max)[b][h] (padded)
  __shared__ float w_s[NB * PS];         // W[a][b][p][k]
  __shared__ float t_s[NB * PS];         // t[b][e] = sum_h c*pose
  __shared__ float u_s[NB * PS];         // u[b][e] = W^T p_out
  __shared__ float pout_s[NB * PS];      // p_out[b][e]
  __shared__ float part_s[2 * 4 * 256];  // GEMM1 partials, column-major [n][m] tiles
  __shared__ float red_s[8 * 32];        // softmax partial max/sum
  __shared__ float colmax_s[NB];
  __shared__ float rcs_s[NB];            // 1/colsum

  // ---- TDM: DMA the (256 x 16) f32 pose tile (row stride 512 floats) into LDS ----
  if (wave == 0) {
    union { uint32_t dw[4]; u32x4 v; } g0;
    union { uint32_t dw[8]; u32x8 v; } g1;
    const uint64_t ga  = (uint64_t)(uintptr_t)(x + (size_t)n * HW * 512 + (size_t)a * PS);
    const uint32_t lds = (uint32_t)(uintptr_t)pose_s;  // low 32 bits = LDS byte offset
    g0.dw[0] = 1u;                                          // count=1, user mode
    g0.dw[1] = lds;                                         // lds_addr
    g0.dw[2] = (uint32_t)ga;                                // global_addr[31:0]
    g0.dw[3] = (uint32_t)((ga >> 32) & 0x1FFFFFFu) | (2u << 30); // addr[56:32], type=2
    g1.dw[0] = 2u << 16;                // data_size=2 (4 bytes); no mask/pad/iterate
    g1.dw[1] = 16u << 16;               // tensor_dim0[15:0]=16 (atomic addr = 0)
    g1.dw[2] = (256u << 16);            // tensor_dim0 hi=0 ; tensor_dim1[15:0]=256
    g1.dw[3] = 16u << 16;               // tensor_dim1 hi=0 ; tile_dim0=16
    g1.dw[4] = 256u;                    // tile_dim1=256 ; tile_dim2=0
    g1.dw[5] = 512u;                    // tensor_dim0_stride[31:0]=512 elements
    g1.dw[6] = 0u;                      // stride hi ; tensor_dim1_stride lo
    g1.dw[7] = 0u;
    asm volatile("tensor_load_to_lds %0, %1" :: "s"(g0.v), "s"(g1.v) : "memory");
    asm volatile("s_wait_tensorcnt 0x0" ::: "memory");
  }
  // ---- load W[a]: 512 floats, layout [b][p][k] ----
  if (tid < 128) {
    ((float4*)w_s)[tid] = ((const float4*)(w + (size_t)a * 512))[tid];
  }
  // ---- init logits to 0 (overlaps with TDM on other waves) ----
  for (int i = tid; i < NB * LTS; i += 256) logitT_s[i] = 0.0f;
  __syncthreads();

  // ---- build transposed pose copy poseT[e][h] from pose[h][e] ----
  {
    float4 r0 = ((const float4*)(pose_s + tid * PS))[0];
    float4 r1 = ((const float4*)(pose_s + tid * PS))[1];
    float4 r2 = ((const float4*)(pose_s + tid * PS))[2];
    float4 r3 = ((const float4*)(pose_s + tid * PS))[3];
    poseT_s[ 0 * PTS + tid] = r0.x; poseT_s[ 1 * PTS + tid] = r0.y;
    poseT_s[ 2 * PTS + tid] = r0.z; poseT_s[ 3 * PTS + tid] = r0.w;
    poseT_s[ 4 * PTS + tid] = r1.x; poseT_s[ 5 * PTS + tid] = r1.y;
    poseT_s[ 6 * PTS + tid] = r1.z; poseT_s[ 7 * PTS + tid] = r1.w;
    poseT_s[ 8 * PTS + tid] = r2.x; poseT_s[ 9 * PTS + tid] = r2.y;
    poseT_s[10 * PTS + tid] = r2.z; poseT_s[11 * PTS + tid] = r2.w;
    poseT_s[12 * PTS + tid] = r3.x; poseT_s[13 * PTS + tid] = r3.y;
    poseT_s[14 * PTS + tid] = r3.z; poseT_s[15 * PTS + tid] = r3.w;
  }
  __syncthreads();

  for (int it = 0; it < NITR; ++it) {
    // =========== softmax over h per column b (rows of logitT) ===========
    {
      const int b = tid & 31;
      const int chunk = tid >> 5;   // 8 chunks of 32 h each
      const float* lrow = logitT_s + b * LTS + chunk * 32;
      float m = -1e30f;
      #pragma unroll
      for (int j = 0; j < 16; ++j) {
        v2f lv = *(const v2f*)(lrow + 2 * j);
        m = fmaxf(m, fmaxf(lv.x, lv.y));
      }
      red_s[chunk * 32 + b] = m;
      __syncthreads();
      if (tid < 32) {
        float mm = red_s[tid];
        for (int j = 1; j < 8; ++j) mm = fmaxf(mm, red_s[j * 32 + tid]);
        colmax_s[tid] = mm;
      }
      __syncthreads();
      const float cm = colmax_s[b];
      float* erow = et_s + b * ETS + chunk * 32;
      float ls = 0.0f;
      #pragma unroll
      for (int j = 0; j < 16; ++j) {
        v2f lv = *(const v2f*)(lrow + 2 * j);
        v2f ev;
        ev.x = __expf(lv.x - cm);
        ev.y = __expf(lv.y - cm);
        *(v2f*)(erow + 2 * j) = ev;
        ls += ev.x + ev.y;
      }
      red_s[chunk * 32 + b] = ls;
      __syncthreads();
      if (tid < 32) {
        float s = 0.0f;
        for (int j = 0; j < 8; ++j) s += red_s[j * 32 + tid];
        rcs_s[tid] = 1.0f / s;
      }
      __syncthreads();
    }

    // ====== GEMM1: t[b,e] = sum_h E[h,b]*pose[h,e]  (M=32,N=16,K=256) ======
    // wave -> (mtile, kgroup); 16 K-steps of K=4, pipelined in batches of 8
    {
      const int mtile = wave & 1;
      const int kgrp  = wave >> 1;
      const float* arow = et_s + (mtile * 16 + lm) * ETS + lh * 2;
      const float* brow = poseT_s + lm * PTS + lh * 2;
      v8f acc = {0.f,0.f,0.f,0.f,0.f,0.f,0.f,0.f};
      for (int half = 0; half < 2; ++half) {
        const int kb0 = kgrp * 64 + half * 32;
        v2f af[8], bf[8];
        #pragma unroll
        for (int j = 0; j < 8; ++j) {
          af[j] = *(const v2f*)(arow + kb0 + j * 4);
          bf[j] = *(const v2f*)(brow + kb0 + j * 4);
        }
        #pragma unroll
        for (int j = 0; j < 8; ++j)
          acc = __builtin_amdgcn_wmma_f32_16x16x4_f32(false, af[j], false, bf[j],
                                                      (short)0, acc, false, false);
      }
      // write partial tile column-major: part[(mtile,kgrp)][n][m]
      float* p = part_s + (mtile * 4 + kgrp) * 256 + lm * 16 + 8 * lh;
      #pragma unroll
      for (int i2 = 0; i2 < 4; ++i2) {
        v2f pv; pv.x = acc[2 * i2]; pv.y = acc[2 * i2 + 1];
        *(v2f*)(p + 2 * i2) = pv;
      }
    }
    __syncthreads();
    // reduce 4 K-group partials per mtile, fold softmax normalization into t
    for (int rep = 0; rep < 2; ++rep) {
      const int idx = rep * 256 + tid;            // 0..511
      const int m     = idx & 15;
      const int nn    = (idx >> 4) & 15;
      const int mtile = idx >> 8;
      float v = 0.0f;
      for (int kg = 0; kg < 4; ++kg)
        v += part_s[(mtile * 4 + kg) * 256 + nn * 16 + m];
      const int b = mtile * 16 + m;
      t_s[b * PS + nn] = v * rcs_s[b];
    }
    __syncthreads();

    // ====== per-b: s = W[b] @ t[b]; squash; u = W[b]^T @ p_out ======
    if (tid < NB) {
      const int b = tid;
      float s[16];
      float n2 = 0.f;
      #pragma unroll
      for (int p = 0; p < 4; ++p)
        #pragma unroll
        for (int q = 0; q < 4; ++q) {
          float acc = 0.f;
          #pragma unroll
          for (int k = 0; k < 4; ++k)
            acc += w_s[b * 16 + p * 4 + k] * t_s[b * 16 + k * 4 + q];
          s[p * 4 + q] = acc;
          n2 += acc * acc;
        }
      const float f = (n2 / (1.0f + n2)) * rsqrtf(n2 + 1e-8f);
      #pragma unroll
      for (int i = 0; i < 16; ++i) pout_s[b * 16 + i] = s[i] * f;
      #pragma unroll
      for (int k = 0; k < 4; ++k)
        #pragma unroll
        for (int q = 0; q < 4; ++q) {
          float acc = 0.f;
          #pragma unroll
          for (int p = 0; p < 4; ++p)
            acc += w_s[b * 16 + p * 4 + k] * pout_s[b * 16 + p * 4 + q];
          u_s[b * 16 + k * 4 + q] = acc;
        }
    }
    __syncthreads();

    // ====== GEMM2: logits[h,b] += sum_e pose[h,e]*u[b,e]  (M=256,N=32,K=16) ======
    if (it < NITR - 1) {
      // hoist B fragments (u) once per wave; reuse A fragments across both n-tiles
      v2f bf0[4], bf1[4];
      #pragma unroll
      for (int kk = 0; kk < 4; ++kk) {
        bf0[kk] = *(const v2f*)(u_s + lm * PS + kk * 4 + lh * 2);
        bf1[kk] = *(const v2f*)(u_s + (16 + lm) * PS + kk * 4 + lh * 2);
      }
      for (int ht2 = 0; ht2 < 2; ++ht2) {
        const int htile = wave * 2 + ht2;
        v2f af[4];
        #pragma unroll
        for (int kk = 0; kk < 4; ++kk)
          af[kk] = *(const v2f*)(pose_s + (htile * 16 + lm) * PS + kk * 4 + lh * 2);
        v8f acc0 = {0.f,0.f,0.f,0.f,0.f,0.f,0.f,0.f};
        v8f acc1 = {0.f,0.f,0.f,0.f,0.f,0.f,0.f,0.f};
        #pragma unroll
        for (int kk = 0; kk < 4; ++kk)
          acc0 = __builtin_amdgcn_wmma_f32_16x16x4_f32(false, af[kk], false, bf0[kk],
                                                       (short)0, acc0, false, false);
        #pragma unroll
        for (int kk = 0; kk < 4; ++kk)
          acc1 = __builtin_amdgcn_wmma_f32_16x16x4_f32(false, af[kk], false, bf1[kk],
                                                       (short)0, acc1, false, false);
        // RMW into logitT: lane column b, rows h = htile*16 + 8*lh + i (contiguous)
        float* l0 = logitT_s + (lm)      * LTS + htile * 16 + 8 * lh;
        float* l1 = logitT_s + (16 + lm) * LTS + htile * 16 + 8 * lh;
        #pragma unroll
        for (int i2 = 0; i2 < 4; ++i2) {
          v2f o = *(v2f*)(l0 + 2 * i2);
          o.x += acc0[2 * i2]; o.y += acc0[2 * i2 + 1];
          *(v2f*)(l0 + 2 * i2) = o;
        }
        #pragma unroll
        for (int i2 = 0; i2 < 4; ++i2) {
          v2f o = *(v2f*)(l1 + 2 * i2);
          o.x += acc1[2 * i2]; o.y += acc1[2 * i2 + 1];
          *(v2f*)(l1 + 2 * i2) = o;
        }
      }
      __syncthreads();
    }
  }

  // ---- write out[n, a, b, e] (pout_s is already [b][e] contiguous) ----
  {
    float2* dst = (float2*)(out + (size_t)(n * NA + a) * (NB * PS));
    const float2* src = (const float2*)pout_s;
    dst[tid] = src[tid];
  }
}

extern "C" void kernel_launch(void* const* d_in, const int* in_sizes, int n_in,
                              void* d_out, int out_size, void* d_ws, size_t ws_size,
                              hipStream_t stream) {
  (void)in_sizes; (void)n_in; (void)out_size; (void)d_ws; (void)ws_size;
  const float* x = (const float*)d_in[0];   // (16,16,16,512) f32
  const float* w = (const float*)d_in[1];   // (32,32,4,4)   f32
  float* out = (float*)d_out;               // (16,32,32,16) f32
  caps_route_kernel<<<dim3(16 * 32), dim3(256), 0, stream>>>(x, w, out);
}